// ContourletTransform_1022202216668
// MI455X (gfx1250) — compile-verified
//
#include <hip/hip_runtime.h>

typedef __attribute__((ext_vector_type(2))) float v2f;
typedef __attribute__((ext_vector_type(4))) float v4f;
typedef __attribute__((ext_vector_type(8))) float v8f;

#define IMG_W   512
#define LL0_W   128
#define PER_CH  9681   // 9555 (scale 0) + 3*42 (scales 1-3)

// One workgroup per (b,c) image. 256 threads = 8 wave32.
__global__ __launch_bounds__(256) void contourlet_fused_kernel(
    const float* __restrict__ x,   // (16,32,512,512) f32
    const float* __restrict__ h,   // (6,2) f32
    const float* __restrict__ g,   // (6,2) f32
    float* __restrict__ out)       // (16,32,9681) f32
{
    // LDS: 64K + 16K + 3*16K + 8K = 136 KB  (320 KB/WGP available -> 2 WG/WGP)
    __shared__ float sA [128 * 128];   // LL ping (levels s=128,32,8 live here)
    __shared__ float sB [64 * 64];     // LL pong (levels s=64,16,4)
    __shared__ float sLH[64 * 64];     // band scratch (stride = hs)
    __shared__ float sHL[64 * 64];
    __shared__ float sHH[64 * 64];
    __shared__ float sT [8 * 256];     // per-wave 16x16 T = W*X scratch

    const int   tid  = threadIdx.x;
    const int   img  = blockIdx.x;           // b*32 + c
    const int   wid  = tid >> 5;             // wave id 0..7 (wave32)
    const int   lane = tid & 31;
    const int   li   = lane & 15;
    const int   half = lane >> 4;
    float*      obase = out + (size_t)img * PER_CH;

    // ---------------- Stage 1: fused dwt2(c,c) x2, LL only: 512 -> 128 ----------------
    // Two Haar-LL levels with c=1/sqrt(2) collapse to 0.25 * sum of each 4x4 block.
    // Input is single-pass (512 MB > 192 MB L2): stream it non-temporally.
    {
        const v4f* xi = (const v4f*)(x + (size_t)img * (IMG_W * IMG_W));
        for (int e = tid; e < LL0_W * LL0_W; e += 256) {
            const int r  = e >> 7;       // output row 0..127
            const int c4 = e & 127;      // output col == float4 col index
            float acc = 0.f;
            #pragma unroll
            for (int k = 0; k < 4; ++k) {
                const v4f v = __builtin_nontemporal_load(&xi[(4 * r + k) * 128 + c4]);
                acc += (v.x + v.y) + (v.z + v.w);
            }
            sA[e] = 0.25f * acc;
        }
    }
    __syncthreads();

    const int off0[6] = {0, 7168, 8960, 9408, 9520, 9548};

    // ---------------- Scale 0: six fused levels ----------------
    for (int l = 0; l < 6; ++l) {
        const int   s  = 128 >> l;        // LL size entering this level
        const int   hs = s >> 1;          // band size after dwt2(h)
        float*      cur = (l & 1) ? sB : sA;
        float*      nxt = (l & 1) ? sA : sB;
        const float f0 = h[2 * l], f1 = h[2 * l + 1];
        const float g0 = g[2 * l], g1 = g[2 * l + 1];

        // ---- Stage A: dwt2(cur, h) -> nxt(LL), sLH, sHL, sHH ----
        if (s >= 16) {
            // WMMA path: per 16x16 tile X, compute Y = W * X * W^T with
            // V_WMMA_F32_16X16X4_F32 (4 K-slices per product).
            // W rows 0..7:  W[m][2m]=f1,   W[m][2m+1]=f0
            // W rows 8..15: W[m][2m']=f0,  W[m][2m'+1]=-f1  (m'=m-8)
            // Quadrants: Y[0:8,0:8]=LL, Y[0:8,8:]=HL, Y[8:,0:8]=LH, Y[8:,8:]=HH
            const int   tpr = s >> 4;
            const int   ntiles = tpr * tpr;
            const int   lim = li & 7;
            const float e0 = (li < 8) ? f1 : f0;
            const float e1 = (li < 8) ? f0 : -f1;
            float wa0[4], wa1[4];          // A-frag of W (also B-frag of W^T)
            #pragma unroll
            for (int kb = 0; kb < 4; ++kb) {
                const int  k0  = 4 * kb + 2 * half;  // A layout: lanes16-31 hold K+2
                const bool hit = (k0 == 2 * lim);    // W row li nonzero at cols {2*lim,2*lim+1}
                wa0[kb] = hit ? e0 : 0.f;
                wa1[kb] = hit ? e1 : 0.f;
            }
            float* Tt = sT + wid * 256;

            for (int t = wid; t < ntiles; t += 8) {   // wave-uniform: EXEC all-1 at WMMA
                const int tr = t / tpr, tc = t - tr * tpr;
                const float* Xt = cur + (tr * 16) * s + tc * 16;

                // T = W * X
                v8f acc = {0, 0, 0, 0, 0, 0, 0, 0};
                #pragma unroll
                for (int kb = 0; kb < 4; ++kb) {
                    const int k0 = 4 * kb + 2 * half;
                    v2f a; a.x = wa0[kb]; a.y = wa1[kb];
                    v2f b; b.x = Xt[k0 * s + li]; b.y = Xt[(k0 + 1) * s + li];
                    acc = __builtin_amdgcn_wmma_f32_16x16x4_f32(
                        false, a, false, b, (short)0, acc, false, false);
                }
                // D layout -> LDS (element (m = v + 8*half, n = li))
                #pragma unroll
                for (int v = 0; v < 8; ++v)
                    Tt[(v + 8 * half) * 16 + li] = acc[v];
                asm volatile("s_wait_dscnt 0" ::: "memory"); // intra-wave cross-lane DS dep

                // Y = T * W^T   (A-frag from Tt, B-frag of W^T == wa pairs)
                v8f y = {0, 0, 0, 0, 0, 0, 0, 0};
                #pragma unroll
                for (int kb = 0; kb < 4; ++kb) {
                    const int k0 = 4 * kb + 2 * half;
                    v2f a2; a2.x = Tt[li * 16 + k0]; a2.y = Tt[li * 16 + k0 + 1];
                    v2f b2; b2.x = wa0[kb]; b2.y = wa1[kb];
                    y = __builtin_amdgcn_wmma_f32_16x16x4_f32(
                        false, a2, false, b2, (short)0, y, false, false);
                }
                // Scatter quadrants: lane holds (m = v + 8*half, n = li)
                const int r8 = tr * 8, c8 = tc * 8;
                if (half == 0) {
                    if (li < 8) {       // LL
                        #pragma unroll
                        for (int v = 0; v < 8; ++v) nxt[(r8 + v) * hs + c8 + li] = y[v];
                    } else {            // HL
                        #pragma unroll
                        for (int v = 0; v < 8; ++v) sHL[(r8 + v) * hs + c8 + (li - 8)] = y[v];
                    }
                } else {
                    if (li < 8) {       // LH
                        #pragma unroll
                        for (int v = 0; v < 8; ++v) sLH[(r8 + v) * hs + c8 + li] = y[v];
                    } else {            // HH
                        #pragma unroll
                        for (int v = 0; v < 8; ++v) sHH[(r8 + v) * hs + c8 + (li - 8)] = y[v];
                    }
                }
                asm volatile("" ::: "memory"); // keep Tt WAR ordering across iterations
            }
        } else {
            // Scalar 2x2-local path for tiny levels (s = 8, 4)
            const int nblk = hs * hs;
            for (int idx = tid; idx < nblk; idx += 256) {
                const int i = idx / hs, j = idx - i * hs;
                const float x00 = cur[(2 * i) * s + 2 * j];
                const float x01 = cur[(2 * i) * s + 2 * j + 1];
                const float x10 = cur[(2 * i + 1) * s + 2 * j];
                const float x11 = cur[(2 * i + 1) * s + 2 * j + 1];
                const float lo0 = f1 * x00 + f0 * x01, hi0 = f0 * x00 - f1 * x01;
                const float lo1 = f1 * x10 + f0 * x11, hi1 = f0 * x10 - f1 * x11;
                nxt[idx] = f1 * lo0 + f0 * lo1;   // LL
                sLH[idx] = f0 * lo0 - f1 * lo1;   // LH
                sHL[idx] = f1 * hi0 + f0 * hi1;   // HL
                sHH[idx] = f0 * hi0 - f1 * hi1;   // HH
            }
        }
        __syncthreads();

        // ---- Stage B: dwt2(band, g); emit A1,H1,V1,A2,H2,H3,D3 to global ----
        {
            const int q  = hs >> 1;
            const int q2 = q * q;
            float* o = obase + off0[l];
            for (int idx = tid; idx < 3 * q2; idx += 256) {
                const int band = idx / q2;
                const int r    = idx - band * q2;
                const int i = r / q, j = r - i * q;
                const float* bb = (band == 0) ? sLH : (band == 1) ? sHL : sHH;
                const float x00 = bb[(2 * i) * hs + 2 * j];
                const float x01 = bb[(2 * i) * hs + 2 * j + 1];
                const float x10 = bb[(2 * i + 1) * hs + 2 * j];
                const float x11 = bb[(2 * i + 1) * hs + 2 * j + 1];
                const float lo0 = g1 * x00 + g0 * x01, hi0 = g0 * x00 - g1 * x01;
                const float lo1 = g1 * x10 + g0 * x11, hi1 = g0 * x10 - g1 * x11;
                const float A = g1 * lo0 + g0 * lo1;  // ll
                const float H = g0 * lo0 - g1 * lo1;  // lh
                const float V = g1 * hi0 + g0 * hi1;  // hl
                const float D = g0 * hi0 - g1 * hi1;  // hh
                if (band == 0) {        // from LH: A1,H1,V1
                    __builtin_nontemporal_store(A, &o[0 * q2 + r]);
                    __builtin_nontemporal_store(H, &o[1 * q2 + r]);
                    __builtin_nontemporal_store(V, &o[2 * q2 + r]);
                } else if (band == 1) { // from HL: A2,H2
                    __builtin_nontemporal_store(A, &o[3 * q2 + r]);
                    __builtin_nontemporal_store(H, &o[4 * q2 + r]);
                } else {                // from HH: H3,D3
                    __builtin_nontemporal_store(H, &o[5 * q2 + r]);
                    __builtin_nontemporal_store(D, &o[6 * q2 + r]);
                }
            }
        }
        __syncthreads();
    }

    // ---------------- Scales 1..3: 2x2 -> 1x1 tail (42 outputs per scale) ----------------
    if (tid == 0) {
        // LL (2x2) sits in sA after level 5
        const float a00 = sA[0], a01 = sA[1], a10 = sA[2], a11 = sA[3];
        const float cst = 0.70710678118654752440f;
        const float lo0 = cst * (a00 + a01), lo1 = cst * (a10 + a11);
        float v = cst * (lo0 + lo1);          // scale-1 leading dwt2 (2->1, even case)
        for (int sc = 1; sc <= 3; ++sc) {
            if (sc > 1) v = 2.0f * v;         // 1x1 dwt2 with c: (c+c)^2 = 2
            for (int l = 0; l < 6; ++l) {
                const float f0 = h[2 * l], f1 = h[2 * l + 1];
                const float g0 = g[2 * l], g1 = g[2 * l + 1];
                const float ph = f0 + f1, mh = f0 - f1;     // 1x1: lo=(f0+f1)x, hi=(f0-f1)x
                const float LLn = ph * ph * v;
                const float LH  = mh * ph * v;
                const float HL  = ph * mh * v;
                const float HH  = mh * mh * v;
                const float pg = g0 + g1, mg = g0 - g1;
                float* o = obase + 9555 + (sc - 1) * 42 + l * 7;
                o[0] = pg * pg * LH;  // A1
                o[1] = mg * pg * LH;  // H1
                o[2] = pg * mg * LH;  // V1
                o[3] = pg * pg * HL;  // A2
                o[4] = mg * pg * HL;  // H2
                o[5] = mg * pg * HH;  // H3
                o[6] = mg * mg * HH;  // D3
                v = LLn;
            }
        }
    }
}

extern "C" void kernel_launch(void* const* d_in, const int* in_sizes, int n_in,
                              void* d_out, int out_size, void* d_ws, size_t ws_size,
                              hipStream_t stream) {
    const float* x = (const float*)d_in[0];   // (16,32,512,512) f32
    const float* h = (const float*)d_in[1];   // (6,2) f32
    const float* g = (const float*)d_in[2];   // (6,2) f32
    float* out = (float*)d_out;               // (16,32,9681) f32
    const int nimg = in_sizes[0] / (IMG_W * IMG_W);   // 512 images
    contourlet_fused_kernel<<<nimg, 256, 0, stream>>>(x, h, g, out);
}